// res_GAT_up_5738076307729
// MI455X (gfx1250) — compile-verified
//
#include <hip/hip_runtime.h>

typedef float v2f __attribute__((ext_vector_type(2)));
typedef float v8f __attribute__((ext_vector_type(8)));

#define BB    8
#define CC_   128
#define NN    4096
#define KNN_  8
#define MT_   (NN / 16)
#define LRELU 0.01f

__device__ __forceinline__ float leaky(float x) { return x > 0.f ? x : LRELU * x; }

// Low 32 bits of a generic pointer to an LDS object == LDS byte offset
// (generic LDS addresses are {shared_aperture_hi32, lds_offset32}).
__device__ __forceinline__ uint32_t lds_off(const void* p) {
  return (uint32_t)(uintptr_t)p;
}

__device__ __forceinline__ void async_b128(uint32_t lds, const void* g) {
  uint64_t ga = (uint64_t)(uintptr_t)g;
  asm volatile("global_load_async_to_lds_b128 %0, %1, off" :: "v"(lds), "v"(ga) : "memory");
}
__device__ __forceinline__ void async_b32(uint32_t lds, const void* g) {
  uint64_t ga = (uint64_t)(uintptr_t)g;
  asm volatile("global_load_async_to_lds_b32 %0, %1, off" :: "v"(lds), "v"(ga) : "memory");
}

// ---------------------------------------------------------------------------
// Kernel 0: xx[b][n] = sum_c points[b][c][n]^2
// ---------------------------------------------------------------------------
__global__ __launch_bounds__(256) void xx_kernel(const float* __restrict__ pts,
                                                 float* __restrict__ xx) {
  int gid = blockIdx.x * 256 + threadIdx.x;         // B*N threads
  int b = gid >> 12;
  int n = gid & (NN - 1);
  const float* p = pts + (size_t)b * CC_ * NN + n;
  float s = 0.f;
#pragma unroll 8
  for (int c = 0; c < CC_; ++c) {
    float v = p[(size_t)c * NN];
    s += v * v;
  }
  xx[gid] = s;
}

// ---------------------------------------------------------------------------
// Kernel 1: KNN via f32 WMMA Gram tiles.
// Block = 4 waves, each owning a 16-row tile; the 16-col B tile is shared by
// all waves, so it is staged once per block into LDS with double-buffered
// global_load_async_to_lds_b128 (ASYNCcnt pipelined: wait<=4 keeps the next
// tile in flight while WMMAs consume the current one).
// ---------------------------------------------------------------------------
__global__ __launch_bounds__(128) void knn_kernel(const float* __restrict__ pts,
                                                  const float* __restrict__ xx,
                                                  int* __restrict__ idx_out) {
  __shared__ float btile[2][CC_ * 16];              // 2 x 8KB double buffer
  __shared__ float tile[4][16 * 16];                // wave-private merge tiles
  const int tid  = threadIdx.x;
  const int wave = tid >> 5;
  const int lane = tid & 31;
  const int col  = lane & 15;
  const int hi   = lane >> 4;                       // K-pair select for A/B
  const int b    = blockIdx.y;
  const int n0   = (blockIdx.x * 4 + wave) * 16;    // grid.x = N/64

  const float* pb = pts + (size_t)b * CC_ * NN;
  const float* xb = xx + (size_t)b * NN;

  // Stage tile 0 immediately (overlaps with A preload below).
  // 2048 floats = 512 b128 transfers; 4 per thread.
  {
    uint32_t lb = lds_off(&btile[0][0]);
#pragma unroll
    for (int i = 0; i < 4; ++i) {
      int t = tid + 128 * i;                        // 0..511
      int c = t >> 2, g4 = (t & 3) * 4;
      async_b128(lb + (uint32_t)(c * 16 + g4) * 4u, pb + (size_t)c * NN + g4);
    }
  }

  // Preload A operand: rows n0..n0+15 over all C (invariant across m-tiles).
  v2f a[32];
#pragma unroll
  for (int s = 0; s < 32; ++s) {
    int c0 = s * 4 + 2 * hi;
    a[s].x = pb[(size_t)c0 * NN + n0 + col];
    a[s].y = pb[(size_t)(c0 + 1) * NN + n0 + col];
  }
  float xr[8];
#pragma unroll
  for (int v = 0; v < 8; ++v) xr[v] = xb[n0 + v + 8 * hi];

  float best_d[KNN_];
  int   best_i[KNN_];
#pragma unroll
  for (int q = 0; q < KNN_; ++q) { best_d[q] = -1e30f; best_i[q] = 0; }

  for (int mt = 0; mt < MT_; ++mt) {
    const int m0  = mt * 16;
    const int cur = mt & 1;
    if (mt + 1 < MT_) {                             // stage next tile into other buffer
      uint32_t lb = lds_off(&btile[cur ^ 1][0]);
      const float* src = pb + (size_t)(m0 + 16);
#pragma unroll
      for (int i = 0; i < 4; ++i) {
        int t = tid + 128 * i;
        int c = t >> 2, g4 = (t & 3) * 4;
        async_b128(lb + (uint32_t)(c * 16 + g4) * 4u, src + (size_t)c * NN + g4);
      }
      asm volatile("s_wait_asynccnt 0x4" ::: "memory");   // current tile done
    } else {
      asm volatile("s_wait_asynccnt 0x0" ::: "memory");
    }
    __syncthreads();                                // tile[cur] visible to all waves

    const float* bt = &btile[cur][0];
    v8f acc0 = {}, acc1 = {};
#pragma unroll
    for (int s = 0; s < 32; s += 2) {               // two independent WMMA chains
      int c0 = s * 4 + 2 * hi;
      int c1 = c0 + 4;
      v2f b0, b1;
      b0.x = bt[c0 * 16 + col];       b0.y = bt[(c0 + 1) * 16 + col];
      b1.x = bt[c1 * 16 + col];       b1.y = bt[(c1 + 1) * 16 + col];
      acc0 = __builtin_amdgcn_wmma_f32_16x16x4_f32(false, a[s], false, b0,
                                                   (short)0, acc0, false, false);
      acc1 = __builtin_amdgcn_wmma_f32_16x16x4_f32(false, a[s + 1], false, b1,
                                                   (short)0, acc1, false, false);
    }

    float xc = xb[m0 + col];
    // C/D layout: vgpr v, lane -> (row = v + 8*hi, col)
#pragma unroll
    for (int v = 0; v < 8; ++v) {
      float nd = 2.f * (acc0[v] + acc1[v]) - xr[v] - xc;
      tile[wave][(v + 8 * hi) * 16 + col] = nd;
    }
    // Same-wave LDS is in-order: wave-private write->read needs no barrier.
    if (lane < 16) {
#pragma unroll
      for (int j = 0; j < 16; ++j) {
        float d = tile[wave][lane * 16 + j];
        if (d > best_d[KNN_ - 1]) {                 // strict > keeps lowest-index ties
          best_d[KNN_ - 1] = d;
          best_i[KNN_ - 1] = m0 + j;
#pragma unroll
          for (int q = KNN_ - 1; q > 0; --q) {
            if (best_d[q] > best_d[q - 1]) {
              float td = best_d[q]; best_d[q] = best_d[q - 1]; best_d[q - 1] = td;
              int   ti = best_i[q]; best_i[q] = best_i[q - 1]; best_i[q - 1] = ti;
            }
          }
        }
      }
    }
    __syncthreads();                                // btile[cur] re-staged next iter
  }

  if (lane < 16) {
    int n = n0 + lane;
#pragma unroll
    for (int q = 0; q < KNN_; ++q)
      idx_out[((size_t)b * NN + n) * KNN_ + q] = best_i[q];
  }
}

// ---------------------------------------------------------------------------
// Kernel 2: out = (sum_k leaky(x[:,idx_k]) + leaky(x)) / 9 + x
// (mean over axis-2 of a softmax is exactly 1/9: the W/attention path cancels)
// ---------------------------------------------------------------------------
__global__ __launch_bounds__(256) void out_kernel(const float* __restrict__ pts,
                                                  const int* __restrict__ idx,
                                                  float* __restrict__ out) {
  __shared__ int sidx[2][KNN_];
  const int b   = blockIdx.y;
  const int tid = threadIdx.x;
  const int nb  = blockIdx.x * 2;                   // grid.x = N/2
  if (tid < 16)
    sidx[tid >> 3][tid & 7] =
        idx[((size_t)b * NN + nb + (tid >> 3)) * KNN_ + (tid & 7)];
  __syncthreads();
  const int c = tid & 127;
  const int w = tid >> 7;
  const int n = nb + w;
  const float* pc = pts + ((size_t)b * CC_ + c) * NN;
  float ctr = pc[n];
  float s = leaky(ctr);
#pragma unroll
  for (int q = 0; q < KNN_; ++q) s += leaky(pc[sidx[w][q]]);
  out[((size_t)b * CC_ + c) * NN + n] = s * (1.f / 9.f) + ctr;
}

// ---------------------------------------------------------------------------
// Kernel 3: grouped_points_nn[b,o,n,k] = sum_c w2[o,c]*leaky(x[b,c,idx[b,n,k]])
// Gather 16 columns via async b32 (per-lane global->LDS gather, swizzled so
// each WMMA K-pair is contiguous -> ds_read_b64), then 8x32 f32 WMMAs with
// dual accumulation chains. leaky() is applied to the B operand, not A:
// leaky is linear per input sign, but must be applied to the gathered data,
// so it is applied after the LDS round-trip on the B reads.
// ---------------------------------------------------------------------------
__global__ __launch_bounds__(128) void conv2_kernel(const float* __restrict__ pts,
                                                    const float* __restrict__ w2,
                                                    const int* __restrict__ idx,
                                                    float* __restrict__ out2) {
  __shared__ float gf[4][CC_ * 16];                 // 4 waves x 8KB
  const int wave = threadIdx.x >> 5;
  const int lane = threadIdx.x & 31;
  const int col  = lane & 15;
  const int hi   = lane >> 4;
  const int b    = blockIdx.y;
  const int n0   = blockIdx.x * 8 + wave * 2;       // grid.x = N/8

  const int n = n0 + (col >> 3);
  const int k = col & 7;
  const int m = idx[((size_t)b * NN + n) * KNN_ + k];
  const float* pb = pts + (size_t)b * CC_ * NN;

  // Async gather of column m: swizzled so (c, c+1) pairs used by one WMMA
  // step land contiguously: sw(c,col) = (((c>>2)*2 + ((c>>1)&1))*16 + col)*2 + (c&1)
  {
    uint32_t lb = lds_off(&gf[wave][0]);
#pragma unroll 4
    for (int i = 0; i < 64; ++i) {
      int c = i * 2 + hi;                           // parity-hi covers all c over lanes
      uint32_t sw = (uint32_t)((((c >> 2) * 2 + ((c >> 1) & 1)) * 16 + col) * 2 + (c & 1));
      async_b32(lb + sw * 4u, pb + (size_t)c * NN + m);
    }
  }
  asm volatile("s_wait_asynccnt 0x0" ::: "memory"); // wave-private: no barrier needed
  const v2f* g = (const v2f*)&gf[wave][0];

#pragma unroll 1
  for (int ot = 0; ot < 8; ++ot) {
    v8f acc0 = {}, acc1 = {};
    const float* wrow = w2 + (size_t)(ot * 16 + col) * CC_;   // A row M = col
#pragma unroll
    for (int s = 0; s < 32; s += 2) {
      int c0 = s * 4 + 2 * hi;
      int c1 = c0 + 4;
      v2f a0, a1;
      a0.x = wrow[c0];     a0.y = wrow[c0 + 1];
      a1.x = wrow[c1];     a1.y = wrow[c1 + 1];
      v2f b0 = g[(s * 2 + hi) * 16 + col];          // ds_read_b64
      v2f b1 = g[((s + 1) * 2 + hi) * 16 + col];
      b0.x = leaky(b0.x);  b0.y = leaky(b0.y);
      b1.x = leaky(b1.x);  b1.y = leaky(b1.y);
      acc0 = __builtin_amdgcn_wmma_f32_16x16x4_f32(false, a0, false, b0,
                                                   (short)0, acc0, false, false);
      acc1 = __builtin_amdgcn_wmma_f32_16x16x4_f32(false, a1, false, b1,
                                                   (short)0, acc1, false, false);
    }
#pragma unroll
    for (int v = 0; v < 8; ++v) {
      int o = ot * 16 + v + 8 * hi;
      out2[(((size_t)b * CC_ + o) * NN + n) * KNN_ + k] = acc0[v] + acc1[v];
    }
  }
}

// ---------------------------------------------------------------------------
extern "C" void kernel_launch(void* const* d_in, const int* in_sizes, int n_in,
                              void* d_out, int out_size, void* d_ws, size_t ws_size,
                              hipStream_t stream) {
  (void)in_sizes; (void)n_in; (void)out_size; (void)ws_size;
  const float* pts = (const float*)d_in[0];
  // d_in[1] = W: algebraically dead (mean of softmax == 1/9).
  const float* w2  = (const float*)d_in[2];

  float* out1 = (float*)d_out;                          // [B,C,N]
  float* out2 = out1 + (size_t)BB * CC_ * NN;           // [B,C,N,K]

  float* xx  = (float*)d_ws;                                          // B*N floats
  int*   idx = (int*)((char*)d_ws + (size_t)BB * NN * sizeof(float)); // B*N*K ints

  hipLaunchKernelGGL(xx_kernel,    dim3(BB * NN / 256), dim3(256), 0, stream, pts, xx);
  hipLaunchKernelGGL(knn_kernel,   dim3(NN / 64, BB),   dim3(128), 0, stream, pts, xx, idx);
  hipLaunchKernelGGL(out_kernel,   dim3(NN / 2, BB),    dim3(256), 0, stream, pts, idx, out1);
  hipLaunchKernelGGL(conv2_kernel, dim3(NN / 8, BB),    dim3(128), 0, stream, pts, w2, idx, out2);
}